// PreEncoder_69157563400693
// MI455X (gfx1250) — compile-verified
//
#include <hip/hip_runtime.h>
#include <stdint.h>

typedef float    v4f  __attribute__((ext_vector_type(4)));
typedef uint32_t u32x4 __attribute__((ext_vector_type(4)));
typedef int      i32x4 __attribute__((ext_vector_type(4)));
typedef int      i32x8 __attribute__((ext_vector_type(8)));

#define ELEMS_PER_BLOCK 512   // 2 KB input tile per block, 32 KB output per block

#if defined(__has_builtin)
#if __has_builtin(__builtin_amdgcn_tensor_load_to_lds) && \
    __has_builtin(__builtin_amdgcn_s_wait_tensorcnt)
#define USE_TDM 1
#endif
#endif
#ifndef USE_TDM
#define USE_TDM 0
#endif

// Output bit j (1..15) of the 16-bit pattern, mapped to +-1.0f.
// g = ~h; bit==1 -> +1.0f, bit==0 -> -1.0f. (g << (16+j)) puts h-bit (15-j) at 31.
__device__ __forceinline__ float bitf(uint32_t g, int j) {
  return __uint_as_float(0x3F800000u | ((g << (16 + j)) & 0x80000000u));
}

__global__ __launch_bounds__(256)
void float2bit_pm1_kernel(const float* __restrict__ x,
                          float* __restrict__ out,
                          int n) {
  const int base = (int)blockIdx.x * ELEMS_PER_BLOCK;

#if USE_TDM
  __shared__ float tile[ELEMS_PER_BLOCK];
  int count = n - base;
  if (count > ELEMS_PER_BLOCK) count = ELEMS_PER_BLOCK;

  if (threadIdx.x < 32u) {  // wave 0 issues the DMA (tensor ops are wave-level)
    // True LDS byte offset of the tile (addrspace(3) pointer value).
    const uint32_t lds_off =
        (uint32_t)(uintptr_t)(__attribute__((address_space(3))) float*)&tile[0];
    const uint64_t gaddr = (uint64_t)(uintptr_t)(x + base);

    // ---- D# group 0: count=1 (valid), lds_addr, global_addr[56:0], type=2 ----
    u32x4 g0;
    g0[0] = 1u;                       // count=1, is_restore=0, gather_mode=0
    g0[1] = lds_off;                  // lds_addr (bytes)
    g0[2] = (uint32_t)gaddr;          // global_addr[31:0]
    g0[3] = (((uint32_t)(gaddr >> 32)) & 0x01FFFFFFu) | (2u << 30);  // [56:32] | type=2

    // ---- D# group 1: wg_mask=0 (not in cluster), data_size=4B, 1-D tile ----
    i32x8 g1;
    g1[0] = (int)(2u << 16);                                // data_size = 2 (4 bytes)
    g1[1] = (int)(((uint32_t)count & 0xFFFFu) << 16);       // tensor_dim0[15:0]
    g1[2] = (int)(1u << 16);                                // tensor_dim0[31:16]=0 | tensor_dim1=1 (lo)
    g1[3] = (int)(((uint32_t)count & 0xFFFFu) << 16);       // tensor_dim1(hi)=0 | tile_dim0=count
    g1[4] = 0;                                              // tile_dim1=0 (1-D), tile_dim2=0
    g1[5] = (int)(uint32_t)count;                           // tensor_dim0_stride[31:0]
    g1[6] = 0;                                              // stride0[47:32] | tensor_dim1_stride(lo)
    g1[7] = 0;                                              // tensor_dim1_stride(hi)

    // ---- D# groups 2/3: higher dims unused (tile_dim3/4 = 0), dims = 1 ----
    i32x4 g2; g2[0] = 1; g2[1] = 1; g2[2] = 0; g2[3] = 0;   // tensor_dim2=1, tensor_dim3=1
    i32x4 g3; g3[0] = 0; g3[1] = (int)(1u << 16); g3[2] = 0; g3[3] = 0;  // tensor_dim4=1

#if __clang_major__ >= 23
    i32x8 g4 = {0, 0, 0, 0, 0, 0, 0, 0};
    __builtin_amdgcn_tensor_load_to_lds(g0, g1, g2, g3, g4, 0);
#else
    __builtin_amdgcn_tensor_load_to_lds(g0, g1, g2, g3, 0);
#endif
    __builtin_amdgcn_s_wait_tensorcnt(0);  // LDS data visible to this wave
  }
  __syncthreads();                          // publish LDS tile to all 8 waves
#endif

#pragma unroll
  for (int k = 0; k < ELEMS_PER_BLOCK / 256; ++k) {
    const int li = (int)threadIdx.x + k * 256;
    const int i  = base + li;
    if (i >= n) break;

#if USE_TDM
    const float f = tile[li];
#else
    const float f = x[i];
#endif
    const uint32_t u = __float_as_uint(f);
    const uint32_t a = u & 0x7fffffffu;   // |f| bits

    // ---------- finite path: build 15-bit (exp<<10 | mant) ----------
    uint32_t a0   = a;
    int      esub = 0;
    if (a0 != 0u && a0 < 0x00800000u) {   // fp32 subnormal: normalize (exact)
      a0   = __float_as_uint(__uint_as_float(a0) * 16777216.0f);  // * 2^24
      esub = 24;
    }
    const int      e_sci  = (int)(a0 >> 23) - 127 - esub;      // floor(log2|f|)
    const uint32_t edec   = (uint32_t)(e_sci + 15) & 31u;      // mod-32 wrap == ref semantics
    const uint32_t mant10 = (a0 >> 13) & 0x3FFu;               // top 10 mantissa bits, truncated
    uint32_t h = (a == 0u) ? 0u : ((edec << 10) | mant10);     // |f|==0 -> all-zero field

    // ---------- sign output: reference computes sign(f + 0.001) ----------
    const float sp    = f + 0.001f;
    float       s_out = (sp < 0.0f) ? 1.0f : ((sp > 0.0f) ? -1.0f : 0.0f);

    // ---------- specials ----------
    if (a > 0x7f800000u) {                 // NaN: 0 11111 1000000000
      h = 0x7E00u; s_out = -1.0f;
    } else if (a == 0x7f800000u) {         // +-Inf: s 11111 0000000000 (true sign)
      h = 0x7C00u; s_out = (u >> 31) ? 1.0f : -1.0f;
    }

    const uint32_t g = ~h;
    v4f o0, o1, o2, o3;
    o0.x = s_out;
    o0.y = bitf(g, 1);   o0.z = bitf(g, 2);   o0.w = bitf(g, 3);
    o1.x = bitf(g, 4);   o1.y = bitf(g, 5);   o1.z = bitf(g, 6);   o1.w = bitf(g, 7);
    o2.x = bitf(g, 8);   o2.y = bitf(g, 9);   o2.z = bitf(g, 10);  o2.w = bitf(g, 11);
    o3.x = bitf(g, 12);  o3.y = bitf(g, 13);  o3.z = bitf(g, 14);  o3.w = bitf(g, 15);

    // 64 B per element as 4 x b128 non-temporal stores; a wave's 4 stores
    // fully tile a contiguous 2 KB region (write-combines in GL2).
    v4f* p = (v4f*)(out + (size_t)i * 16u);
    __builtin_nontemporal_store(o0, p + 0);
    __builtin_nontemporal_store(o1, p + 1);
    __builtin_nontemporal_store(o2, p + 2);
    __builtin_nontemporal_store(o3, p + 3);
  }
}

extern "C" void kernel_launch(void* const* d_in, const int* in_sizes, int n_in,
                              void* d_out, int out_size, void* d_ws, size_t ws_size,
                              hipStream_t stream) {
  const float* x   = (const float*)d_in[0];
  float*       out = (float*)d_out;
  const int    n   = in_sizes[0];              // 256*2048*4 = 2,097,152

  int grid = (n + ELEMS_PER_BLOCK - 1) / ELEMS_PER_BLOCK;  // 4096 blocks
  if (grid < 1) grid = 1;

  float2bit_pm1_kernel<<<grid, 256, 0, stream>>>(x, out, n);
}